// CrossAttention_36309653521078
// MI455X (gfx1250) — compile-verified
//
#include <hip/hip_runtime.h>
#include <math.h>
#include <stdint.h>

// ---------------------------------------------------------------------------
// Sizes from the reference
// ---------------------------------------------------------------------------
constexpr int Bn   = 16384;   // batch
constexpr int SD   = 3072;    // h_s width
constexpr int HID  = 512;     // hidden = H*DK = G_DIM
constexpr int HID2 = 1024;    // MLP inner

// GEMM tiling: block = 64 rows x 512 cols, 16 waves (wave32), wave = 16x128
constexpr int MT   = 64;      // rows per block
constexpr int KT   = 32;      // k-tile staged in LDS (64 WMMAs/wave per stage)
constexpr int PADL = KT + 2;  // LDS A-row pad (even -> 8B-aligned float2 frags)
constexpr int ASZ  = MT * PADL;   // one A buffer (floats)
constexpr int BSZ  = KT * 512;    // one B buffer (floats, row-major [k][n])

typedef __attribute__((ext_vector_type(2))) float v2f;
typedef __attribute__((ext_vector_type(8))) float v8f;
typedef __attribute__((ext_vector_type(4))) int   v4i;
typedef __attribute__((ext_vector_type(4))) unsigned int u32x4;
typedef __attribute__((ext_vector_type(8))) int   i32x8;
typedef __attribute__((ext_vector_type(4))) int   i32x4;

// ---- probe-guarded CDNA5 features -----------------------------------------
#if defined(__has_builtin)
#if __has_builtin(__builtin_amdgcn_global_load_async_to_lds_b128)
#define HAVE_ASYNC_LDS 1
#endif
#if __has_builtin(__builtin_amdgcn_s_wait_asynccnt)
#define HAVE_WAIT_ASYNC 1
#endif
#if __has_builtin(__builtin_amdgcn_tensor_load_to_lds) && \
    __has_builtin(__builtin_amdgcn_s_wait_tensorcnt)
#define HAVE_TDM 1
#endif
#endif

#if defined(HAVE_ASYNC_LDS)
typedef __attribute__((address_space(1))) v4i gvec_t;
typedef __attribute__((address_space(3))) v4i lvec_t;
__device__ __forceinline__ void async_copy16(const float* g, float* lds) {
    __builtin_amdgcn_global_load_async_to_lds_b128(
        (gvec_t*)(float*)g, (lvec_t*)lds, 0, 0);
}
__device__ __forceinline__ void async_wait(int n) {
#if defined(HAVE_WAIT_ASYNC)
    if (n == 0) __builtin_amdgcn_s_wait_asynccnt(0);
    else        __builtin_amdgcn_s_wait_asynccnt(1);
#else
    if (n == 0) asm volatile("s_wait_asynccnt 0x0" ::: "memory");
    else        asm volatile("s_wait_asynccnt 0x1" ::: "memory");
#endif
}
#endif

#if defined(HAVE_TDM)
// Build a D# (ISA 8.3/8.4/8.5) and issue TENSOR_LOAD_TO_LDS.
//   tile: x (contiguous, data_size units) * y (stride s0) * z (stride s1; 0=2D)
//   tensor dims set == tile dims (tiles never cross the tensor edge here)
__device__ __forceinline__ void tdm_load(const float* g, uint32_t lds_off,
                                         int tx, int ty, int tz,
                                         uint64_t s0, uint64_t s1,
                                         uint32_t padbits) {
    const uint64_t ga = (uint64_t)(uintptr_t)g;
    u32x4 g0;
    g0[0] = 1u;                                     // count=1, user mode
    g0[1] = lds_off;                                // lds_addr
    g0[2] = (uint32_t)ga;                           // global_addr[31:0]
    g0[3] = (uint32_t)((ga >> 32) & 0x01FFFFFFu) | (2u << 30);  // addr[56:32], type=2
    i32x8 g1;
    g1[0] = (int)((2u << 16) | padbits);            // data_size=4B (+pad ctl)
    g1[1] = (int)(((uint32_t)tx & 0xFFFFu) << 16);  // tensor_dim0 lo16
    g1[2] = (int)((((uint32_t)tx >> 16) & 0xFFFFu) |
                  (((uint32_t)ty & 0xFFFFu) << 16));            // dim0 hi / dim1 lo
    g1[3] = (int)((((uint32_t)ty >> 16) & 0xFFFFu) |
                  (((uint32_t)tx & 0xFFFFu) << 16));            // dim1 hi / tile_dim0
    g1[4] = (int)(((uint32_t)ty & 0xFFFFu) |
                  (((uint32_t)tz & 0xFFFFu) << 16));            // tile_dim1 / tile_dim2
    g1[5] = (int)(uint32_t)(s0 & 0xFFFFFFFFu);                  // dim0_stride lo32
    g1[6] = (int)((uint32_t)((s0 >> 32) & 0xFFFFu) |
                  ((uint32_t)(s1 & 0xFFFFu) << 16));            // dim0_stride hi / dim1_stride lo
    g1[7] = (int)(uint32_t)((s1 >> 16) & 0xFFFFFFFFu);          // dim1_stride hi32
    i32x4 g2;
    g2[0] = tz; g2[1] = 0; g2[2] = 0; g2[3] = 0;    // tensor_dim2 (0 if 2D)
    i32x4 g3 = (i32x4)0;
#if __clang_major__ >= 23
    i32x8 z8 = (i32x8)0;
    __builtin_amdgcn_tensor_load_to_lds(g0, g1, g2, g3, z8, 0);
#else
    __builtin_amdgcn_tensor_load_to_lds(g0, g1, g2, g3, 0);
#endif
}
#endif

// MODE 0: out = acc + bias                      (V projection, gather-B)
// MODE 1: out = LayerNorm(acc + bias + res)     (attn-out / final)
// MODE 2: out = gelu_exact(acc + bias)          (MLP up-proj)
template <int MODE, bool GATHER_B>
__global__ __launch_bounds__(512)
void gemm_fused(const float* __restrict__ A,     // [Bn, Kdim]
                const float* __restrict__ Bw,    // [Kdim, ldB] or Wv [8,SD,64]
                const float* __restrict__ bias,  // [Nfull]
                const float* __restrict__ res,   // [Bn, 512] (MODE 1)
                const float* __restrict__ lng,   // [512]     (MODE 1)
                const float* __restrict__ lnb,   // [512]     (MODE 1)
                float* __restrict__ out,         // [Bn, ldOut]
                int Kdim, int ldB, int ldOut) {
#if defined(HAVE_TDM)
    __shared__ float lA[2 * ASZ];            // ping-pong A, padded rows [row][KT+2]
    __shared__ float lB[2 * BSZ];            // ping-pong B, row-major [k][512]
#else
    __shared__ float lA[ASZ];                // A tile, row-major [row][k]
    __shared__ float lB[512 * PADL];         // B tile, transposed [col][k]
#endif
    __shared__ float sP[MT][4], sQ[MT][4];   // LN cross-wave partials
    __shared__ float sMu[MT], sRs[MT];

    const int t     = threadIdx.x;           // 0..511
    const int wid   = t >> 5;                // 0..15
    const int L     = t & 31;                // lane (wave32)
    const int waveM = wid >> 2;              // 0..3 -> 16-row span
    const int waveN = wid & 3;               // 0..3 -> 128-col span
    const int cL    = L & 15;
    const int half  = L >> 4;
    const int blockRow = blockIdx.x * MT;
    const int n0       = blockIdx.y * 512;   // only the W1 GEMM has gridDim.y == 2
    const int kb0      = half * 2;           // f32 A/B frag: lanes16-31 hold K={2,3}

    // A-tile staging: MT*KT floats = 512 float4 chunks, one per thread
    const int aRow = t >> 3, aC4 = (t & 7) * 4;

    v8f acc[8];
#pragma unroll
    for (int i = 0; i < 8; ++i) acc[i] = (v8f)0.0f;

#if defined(HAVE_TDM)
    // scalar-provable wave-0 test: TDM ops ignore EXEC, so we need a real branch
    const bool isW0 = (__builtin_amdgcn_readfirstlane(wid) == 0);

    auto stage = [&](int k0s, int b) {
        {                                    // A tile: 512 async 16B chunks
            const float* gp = A + (size_t)(blockRow + aRow) * Kdim + k0s + aC4;
            float* dst = &lA[b * ASZ + aRow * PADL + aC4];
#if defined(HAVE_ASYNC_LDS)
            async_copy16(gp, dst);
#else
            const float4 v = *(const float4*)gp;
            dst[0] = v.x; dst[1] = v.y; dst[2] = v.z; dst[3] = v.w;
#endif
        }
        if (isW0) {                          // B panel: one TDM descriptor
            const uint32_t ldsB = (uint32_t)(uintptr_t)&lB[b * BSZ];
            if (GATHER_B) {
                // Wv[h][k][d] -> LDS [k][h*64+d]: x=d(64), y=h(stride SD*64), z=k(stride 64)
                tdm_load(Bw + (size_t)k0s * 64, ldsB,
                         64, 8, KT, (uint64_t)SD * 64, 64, 0);
            } else {
                // plain row-major panel: x=n(512), y=k(stride ldB)
                tdm_load(Bw + (size_t)k0s * ldB + n0, ldsB,
                         512, KT, 0, (uint64_t)ldB, 0, 0);
            }
        }
    };

    stage(0, 0);
    int buf = 0;
    for (int k0 = 0; k0 < Kdim; k0 += KT) {
        const bool more = (k0 + KT) < Kdim;
        if (more) stage(k0 + KT, buf ^ 1);
        if (isW0) {                           // current panel's DMA retired?
            if (more) __builtin_amdgcn_s_wait_tensorcnt(1);
            else      __builtin_amdgcn_s_wait_tensorcnt(0);
        }
#if defined(HAVE_ASYNC_LDS)
        async_wait(more ? 1 : 0);
#endif
        __syncthreads();                      // all waves see buf's data

        const float* aB = &lA[buf * ASZ + (waveM * 16 + cL) * PADL];
        const float* bB = &lB[buf * BSZ + waveN * 128 + cL];
#pragma unroll
        for (int kk = 0; kk < KT; kk += 4) {
            const int kb = kk + kb0;
            const v2f a = *(const v2f*)(aB + kb);
#pragma unroll
            for (int nt = 0; nt < 8; ++nt) {
                v2f b;
                b.x = bB[kb * 512 + nt * 16];
                b.y = bB[(kb + 1) * 512 + nt * 16];
                acc[nt] = __builtin_amdgcn_wmma_f32_16x16x4_f32(
                    false, a, false, b, (short)0, acc[nt], false, false);
            }
        }
        __syncthreads();                      // done reading buf before reuse
        buf ^= 1;
    }
#else  // ---------------- fallback: synchronous staging ----------------------
    for (int k0 = 0; k0 < Kdim; k0 += KT) {
        __syncthreads();
        {
            const float* gp = A + (size_t)(blockRow + aRow) * Kdim + k0 + aC4;
            float* dst = &lA[aRow * PADL + aC4];
#if defined(HAVE_ASYNC_LDS)
            async_copy16(gp, dst);
#else
            const float4 v = *(const float4*)gp;
            dst[0] = v.x; dst[1] = v.y; dst[2] = v.z; dst[3] = v.w;
#endif
        }
#pragma unroll
        for (int i = 0; i < KT / 4; ++i) {
            const int idx = i * 512 + t;
            const int gk  = idx >> 7;
            const int col = (idx & 127) * 4;
            const int kk  = k0 + gk;
            const float* gp;
            if (GATHER_B) {
                const int n = n0 + col;
                gp = Bw + ((size_t)(n >> 6) * SD + kk) * 64 + (n & 63);
            } else {
                gp = Bw + (size_t)kk * ldB + n0 + col;
            }
            const float4 v = *(const float4*)gp;
            float* dst = &lB[col * PADL + gk];
            dst[0 * PADL] = v.x; dst[1 * PADL] = v.y;
            dst[2 * PADL] = v.z; dst[3 * PADL] = v.w;
        }
#if defined(HAVE_ASYNC_LDS)
        async_wait(0);
#endif
        __syncthreads();
        const float* aBase = &lA[(waveM * 16 + cL) * PADL];
        const float* bBase = &lB[(waveN * 128 + cL) * PADL];
#pragma unroll
        for (int kk = 0; kk < KT; kk += 4) {
            const v2f a = *(const v2f*)(aBase + kk + kb0);
#pragma unroll
            for (int nt = 0; nt < 8; ++nt) {
                const v2f b = *(const v2f*)(bBase + nt * 16 * PADL + kk + kb0);
                acc[nt] = __builtin_amdgcn_wmma_f32_16x16x4_f32(
                    false, a, false, b, (short)0, acc[nt], false, false);
            }
        }
    }
#endif

    // ------------------------------------------------------------------ epilogue
    // C/D layout: VGPR i, lanes 0-15 -> row i, lanes 16-31 -> row i+8; col = lane%16
#pragma unroll
    for (int nt = 0; nt < 8; ++nt) {
        const int n  = n0 + waveN * 128 + nt * 16 + cL;
        const float bv_ = bias[n];
#pragma unroll
        for (int i = 0; i < 8; ++i) {
            const int r = blockRow + waveM * 16 + half * 8 + i;
            float v = acc[nt][i] + bv_;
            if (MODE == 1) v += res[(size_t)r * 512 + (n - n0)];
            if (MODE == 2) v = 0.5f * v * (1.0f + erff(v * 0.70710678118654752f));
            if (MODE == 1) acc[nt][i] = v;                       // keep for LN
            else           out[(size_t)r * ldOut + n] = v;
        }
    }

    if (MODE == 1) {
#pragma unroll
        for (int i = 0; i < 8; ++i) {
            float s = 0.0f, q = 0.0f;
#pragma unroll
            for (int nt = 0; nt < 8; ++nt) { const float v = acc[nt][i]; s += v; q += v * v; }
#pragma unroll
            for (int m = 1; m < 16; m <<= 1) {
                s += __shfl_xor(s, m, 32);
                q += __shfl_xor(q, m, 32);
            }
            if (cL == 0) {
                const int rl = waveM * 16 + half * 8 + i;
                sP[rl][waveN] = s; sQ[rl][waveN] = q;
            }
        }
        __syncthreads();
        if (t < MT) {
            const float s  = sP[t][0] + sP[t][1] + sP[t][2] + sP[t][3];
            const float q  = sQ[t][0] + sQ[t][1] + sQ[t][2] + sQ[t][3];
            const float mu = s * (1.0f / 512.0f);
            const float var = q * (1.0f / 512.0f) - mu * mu;
            sMu[t] = mu;
            sRs[t] = rsqrtf(var + 1e-5f);
        }
        __syncthreads();
#pragma unroll
        for (int nt = 0; nt < 8; ++nt) {
            const int n = waveN * 128 + nt * 16 + cL;
            const float g = lng[n], bb = lnb[n];
#pragma unroll
            for (int i = 0; i < 8; ++i) {
                const int rl = waveM * 16 + half * 8 + i;
                const int r  = blockRow + rl;
                out[(size_t)r * ldOut + n] = (acc[nt][i] - sMu[rl]) * sRs[rl] * g + bb;
            }
        }
    }
}

// ---------------------------------------------------------------------------
extern "C" void kernel_launch(void* const* d_in, const int* in_sizes, int n_in,
                              void* d_out, int out_size, void* d_ws, size_t ws_size,
                              hipStream_t stream) {
    (void)in_sizes; (void)n_in; (void)out_size; (void)ws_size;
    const float* h_g  = (const float*)d_in[0];
    const float* h_s  = (const float*)d_in[1];
    // d_in[2..5] (Wq,bq,Wk,bk) are mathematically dead: softmax over 1 key == 1
    const float* Wv   = (const float*)d_in[6];
    const float* bv   = (const float*)d_in[7];   // [H,DK] flat == [512] in n-order
    const float* Wo   = (const float*)d_in[8];
    const float* bo   = (const float*)d_in[9];
    const float* ln1g = (const float*)d_in[10];
    const float* ln1b = (const float*)d_in[11];
    const float* W1   = (const float*)d_in[12];
    const float* b1   = (const float*)d_in[13];
    const float* W2   = (const float*)d_in[14];
    const float* b2   = (const float*)d_in[15];
    const float* ln2g = (const float*)d_in[16];
    const float* ln2b = (const float*)d_in[17];
    float* out = (float*)d_out;

    float* wsV = (float*)d_ws;                       // [Bn,512]  V (== attn)
    float* wsA = wsV + (size_t)Bn * HID;             // [Bn,512]  h_attn
    float* wsM = wsA + (size_t)Bn * HID;             // [Bn,1024] gelu(mid)

    const dim3 blk(512);
    // 1) V = h_s @ Wv_t + bv          (K=3072, 3D-TDM gather-transpose of Wv)
    gemm_fused<0, true ><<<dim3(Bn / MT, 1), blk, 0, stream>>>(
        h_s, Wv, bv, nullptr, nullptr, nullptr, wsV, SD, HID, HID);
    // 2) h_attn = LN1(V @ Wo + bo + h_g)
    gemm_fused<1, false><<<dim3(Bn / MT, 1), blk, 0, stream>>>(
        wsV, Wo, bo, h_g, ln1g, ln1b, wsA, HID, HID, HID);
    // 3) mid = gelu(h_attn @ W1 + b1)  (N=1024 -> gridDim.y = 2)
    gemm_fused<2, false><<<dim3(Bn / MT, 2), blk, 0, stream>>>(
        wsA, W1, b1, nullptr, nullptr, nullptr, wsM, HID, HID2, HID2);
    // 4) out = LN2(mid @ W2 + b2 + h_attn)
    gemm_fused<1, false><<<dim3(Bn / MT, 1), blk, 0, stream>>>(
        wsM, W2, b2, wsA, ln2g, ln2b, out, HID2, HID, HID);
}